// GResBlockConv_35716948034108
// MI455X (gfx1250) — compile-verified
//
#include <hip/hip_runtime.h>

// ---------------------------------------------------------------------------
// 2-layer GATConv forward for MI455X (gfx1250, wave32).
// GEMMs use V_WMMA_F32_16X16X4_F32 (full-precision f32 matrix core).
// Edge phase uses L2-resident float atomics (segment max / sum / scatter-add),
// with b128 vector loads on the bandwidth-dominant scatter pass.
// ---------------------------------------------------------------------------

#define NEG_SLOPE 0.2f

typedef float v2f __attribute__((ext_vector_type(2)));
typedef float v8f __attribute__((ext_vector_type(8)));

// ---- monotone order-preserving f32 <-> u32 key (for atomicMax segment-max) --
__device__ __forceinline__ unsigned fkey(float f) {
  unsigned u = __float_as_uint(f);
  return (u & 0x80000000u) ? ~u : (u | 0x80000000u);
}
__device__ __forceinline__ float funkey(unsigned k) {
  return (k & 0x80000000u) ? __uint_as_float(k & 0x7FFFFFFFu)
                           : __uint_as_float(~k);
}

__device__ __forceinline__ void edge_sd(const int* __restrict__ ei, int e,
                                        int E, int& s, int& d) {
  if (e < E) { s = ei[e]; d = ei[E + e]; }   // edge_index[0]=src, [1]=dst
  else       { s = d = e - E; }              // appended self loops
}

// ---------------------------------------------------------------------------
// K1: H[nrows,128] = X[nrows,128] @ W[128,128] via v_wmma_f32_16x16x4_f32.
// One wave per 16x16 output tile; 32 WMMAs over K=128.
// nrows must be a multiple of 16 (N=40000 is).
// ---------------------------------------------------------------------------
__global__ void gemm128_wmma(const float* __restrict__ X,
                             const float* __restrict__ W,
                             float* __restrict__ Hh, int nrows) {
  int wid   = blockIdx.x * (blockDim.x >> 5) + (threadIdx.x >> 5);
  int total = (nrows >> 4) << 3;                 // (nrows/16) * (128/16)
  if (wid >= total) return;                      // whole-wave exit only
  int tm = wid >> 3, tn = wid & 7;
  int lane = threadIdx.x & 31;
  int half = lane >> 4;                          // 0: K={0,1}, 1: K={2,3}
  int r    = lane & 15;

  const float* xrow = X + (size_t)(tm * 16 + r) * 128;   // A: M = r per lane
  const float* wcol = W + tn * 16 + r;                   // B: N = r per lane
  v8f c = {};
#pragma unroll 4
  for (int k = 0; k < 128; k += 4) {
    v2f a, b;
    a.x = xrow[k + 2 * half];                    // A 16x4: [r, k+2h+{0,1}]
    a.y = xrow[k + 2 * half + 1];
    b.x = wcol[(size_t)(k + 2 * half) * 128];    // B 4x16: [k+2h+{0,1}, col]
    b.y = wcol[(size_t)(k + 2 * half + 1) * 128];
    c = __builtin_amdgcn_wmma_f32_16x16x4_f32(
        /*neg_a=*/false, a, /*neg_b=*/false, b,
        /*c_mod=*/(short)0, c, /*reuse_a=*/false, /*reuse_b=*/false);
  }
  // D 16x16 f32: VGPR v -> M = v + 8*half, N = r
  float* out = Hh + (size_t)(tm * 16 + 8 * half) * 128 + tn * 16 + r;
#pragma unroll
  for (int v = 0; v < 8; ++v) out[(size_t)v * 128] = c[v];
}

// ---------------------------------------------------------------------------
// K2: zero accumulator (b128) + softmax state for this layer.
// ---------------------------------------------------------------------------
__global__ void init_state(float4* __restrict__ acc4, float* __restrict__ asum,
                           unsigned* __restrict__ amax, int n32, int nH) {
  int i = blockIdx.x * blockDim.x + threadIdx.x;
  if (i < n32) acc4[i] = make_float4(0.f, 0.f, 0.f, 0.f);
  if (i < nH) { asum[i] = 0.0f; amax[i] = 0u; }  // key 0 == -inf sentinel
}

// ---------------------------------------------------------------------------
// K3: per-node attention dots  a_s[i,h] = <h[i,h,:], att_src[h,:]>, same a_d.
// ---------------------------------------------------------------------------
__global__ void node_prep(const float* __restrict__ Hh,
                          const float* __restrict__ att_s,
                          const float* __restrict__ att_d,
                          float* __restrict__ as_, float* __restrict__ ad_,
                          int n, int heads, int C) {
  int i = blockIdx.x * blockDim.x + threadIdx.x;
  if (i >= n * heads) return;
  int node = i / heads, h = i % heads;
  const float* row = Hh + (size_t)node * 128 + h * C;
  const float* vs  = att_s + h * C;
  const float* vd  = att_d + h * C;
  float s = 0.0f, d = 0.0f;
  for (int c = 0; c < C; ++c) { float x = row[c]; s += x * vs[c]; d += x * vd[c]; }
  as_[i] = s; ad_[i] = d;
}

// ---------------------------------------------------------------------------
// K4: edge logits (leaky relu) + segment max via monotone-u32 atomicMax.
// ---------------------------------------------------------------------------
__global__ void edge_logit_max(const int* __restrict__ ei, int E, int Et,
                               int heads, const float* __restrict__ as_,
                               const float* __restrict__ ad_,
                               float* __restrict__ alpha,
                               unsigned* __restrict__ amax) {
  int i = blockIdx.x * blockDim.x + threadIdx.x;
  if (i >= Et * heads) return;
  int e = i / heads, h = i % heads, s, d;
  edge_sd(ei, e, E, s, d);
  float a = as_[s * heads + h] + ad_[d * heads + h];
  a = (a > 0.0f) ? a : NEG_SLOPE * a;
  alpha[i] = a;
  atomicMax(&amax[d * heads + h], fkey(a));
}

// K5: alpha = exp(logit - max[dst]);  atomic segment sum.
__global__ void edge_exp_sum(const int* __restrict__ ei, int E, int Et,
                             int heads, float* __restrict__ alpha,
                             const unsigned* __restrict__ amax,
                             float* __restrict__ asum) {
  int i = blockIdx.x * blockDim.x + threadIdx.x;
  if (i >= Et * heads) return;
  int e = i / heads, h = i % heads, s, d;
  edge_sd(ei, e, E, s, d);
  float ex = __expf(alpha[i] - funkey(amax[d * heads + h]));
  alpha[i] = ex;
  atomicAdd(&asum[d * heads + h], ex);
}

// K6: normalize once per (edge,head) so scatter avoids per-channel divides.
__global__ void edge_norm(const int* __restrict__ ei, int E, int Et, int heads,
                          float* __restrict__ alpha,
                          const float* __restrict__ asum) {
  int i = blockIdx.x * blockDim.x + threadIdx.x;
  if (i >= Et * heads) return;
  int e = i / heads, h = i % heads, s, d;
  edge_sd(ei, e, E, s, d);
  alpha[i] = alpha[i] / (asum[d * heads + h] + 1e-16f);
}

// ---------------------------------------------------------------------------
// K7: scatter  acc[dst, c] += h[src, c] * alpha[e, head(c)].
// Thread per (edge, 4 channels): one global_load_b128 + 4 coalesced f32
// atomics. A 4-channel group never straddles a head boundary (mult. of 64).
// ---------------------------------------------------------------------------
__global__ void edge_scatter(const int* __restrict__ ei, int E, int Et,
                             int heads, int C, const float* __restrict__ Hh,
                             const float* __restrict__ alpha,
                             float* __restrict__ acc) {
  long long i   = (long long)blockIdx.x * blockDim.x + threadIdx.x;
  long long tot = (long long)Et * 32;
  if (i >= tot) return;
  int e  = (int)(i >> 5);
  int c0 = ((int)(i & 31)) << 2;     // channel group base: 0,4,...,124
  int h  = c0 / C;                   // heads=2,C=64 -> c0>>6 ; heads=1 -> 0
  int s, d;
  edge_sd(ei, e, E, s, d);
  float w = alpha[(size_t)e * heads + h];
  const float4 hv = *(const float4*)(Hh + (size_t)s * 128 + c0);
  float* ap = acc + (size_t)d * 128 + c0;
  atomicAdd(ap + 0, hv.x * w);
  atomicAdd(ap + 1, hv.y * w);
  atomicAdd(ap + 2, hv.z * w);
  atomicAdd(ap + 3, hv.w * w);
}

// K8a: layer-1 finalize: x2 = relu(acc + b1)   (b128 in/out)
__global__ void finalize_relu(const float4* __restrict__ acc4,
                              const float4* __restrict__ b4,
                              float4* __restrict__ out4, int n32) {
  int i = blockIdx.x * blockDim.x + threadIdx.x;
  if (i >= n32) return;
  float4 a = acc4[i], b = b4[i & 31], v;
  v.x = fmaxf(a.x + b.x, 0.f);
  v.y = fmaxf(a.y + b.y, 0.f);
  v.z = fmaxf(a.z + b.z, 0.f);
  v.w = fmaxf(a.w + b.w, 0.f);
  out4[i] = v;
}

// K8b: layer-2 finalize: out = relu(acc + b2 + x0)   (residual, b128)
__global__ void finalize_residual(const float4* __restrict__ acc4,
                                  const float4* __restrict__ b4,
                                  const float4* __restrict__ x04,
                                  float4* __restrict__ out4, int n32) {
  int i = blockIdx.x * blockDim.x + threadIdx.x;
  if (i >= n32) return;
  float4 a = acc4[i], b = b4[i & 31], x = x04[i], v;
  v.x = fmaxf(a.x + b.x + x.x, 0.f);
  v.y = fmaxf(a.y + b.y + x.y, 0.f);
  v.z = fmaxf(a.z + b.z + x.z, 0.f);
  v.w = fmaxf(a.w + b.w + x.w, 0.f);
  out4[i] = v;
}

// ---------------------------------------------------------------------------
extern "C" void kernel_launch(void* const* d_in, const int* in_sizes, int n_in,
                              void* d_out, int out_size, void* d_ws,
                              size_t ws_size, hipStream_t stream) {
  const float* x   = (const float*)d_in[0];
  const int*   ei  = (const int*)d_in[1];   // edge_index [2,E] (harness: int)
  // d_in[2] = edge_attr: ignored (GATConv built with edge_dim=None)
  const float* W1  = (const float*)d_in[3];
  const float* as1 = (const float*)d_in[4];
  const float* ad1 = (const float*)d_in[5];
  const float* b1  = (const float*)d_in[6];
  const float* W2  = (const float*)d_in[7];
  const float* as2 = (const float*)d_in[8];
  const float* ad2 = (const float*)d_in[9];
  const float* b2  = (const float*)d_in[10];

  const int n  = in_sizes[0] / 128;   // 40000
  const int E  = in_sizes[1] / 2;     // 640000
  const int Et = E + n;               // + self loops

  // ---- workspace carve-out (all L2-resident: ~67 MB) ----
  float* ws = (float*)d_ws;
  size_t o = 0;
  float*    fH    = ws + o; o += (size_t)n * 128;   // projection h
  float*    fACC  = ws + o; o += (size_t)n * 128;   // aggregation accumulator
  float*    fX2   = ws + o; o += (size_t)n * 128;   // layer-1 output
  float*    fAS   = ws + o; o += (size_t)n * 2;
  float*    fAD   = ws + o; o += (size_t)n * 2;
  float*    fSUM  = ws + o; o += (size_t)n * 2;
  unsigned* uMAX  = (unsigned*)(ws + o); o += (size_t)n * 2;
  float*    fALPH = ws + o; o += (size_t)Et * 2;

  const int gemmBlocks = (((n >> 4) << 3) + 3) / 4;        // 4 waves / block
  const int n32 = n * 32;
  const long long sc = (long long)Et * 32;
  const int scBlocks = (int)((sc + 255) / 256);

  // ======================= Layer 1: heads=2, C=64 =======================
  {
    const int H = 2, C = 64;
    const int nH = n * H, eH = Et * H;
    gemm128_wmma<<<gemmBlocks, 128, 0, stream>>>(x, W1, fH, n);
    init_state<<<(n32 + 255) / 256, 256, 0, stream>>>((float4*)fACC, fSUM, uMAX, n32, nH);
    node_prep<<<(nH + 127) / 128, 128, 0, stream>>>(fH, as1, ad1, fAS, fAD, n, H, C);
    edge_logit_max<<<(eH + 255) / 256, 256, 0, stream>>>(ei, E, Et, H, fAS, fAD, fALPH, uMAX);
    edge_exp_sum<<<(eH + 255) / 256, 256, 0, stream>>>(ei, E, Et, H, fALPH, uMAX, fSUM);
    edge_norm<<<(eH + 255) / 256, 256, 0, stream>>>(ei, E, Et, H, fALPH, fSUM);
    edge_scatter<<<scBlocks, 256, 0, stream>>>(ei, E, Et, H, C, fH, fALPH, fACC);
    finalize_relu<<<(n32 + 255) / 256, 256, 0, stream>>>((const float4*)fACC, (const float4*)b1,
                                                         (float4*)fX2, n32);
  }

  // ====================== Layer 2: heads=1, C=128 =======================
  {
    const int H = 1, C = 128;
    const int nH = n * H, eH = Et * H;
    gemm128_wmma<<<gemmBlocks, 128, 0, stream>>>(fX2, W2, fH, n);
    init_state<<<(n32 + 255) / 256, 256, 0, stream>>>((float4*)fACC, fSUM, uMAX, n32, nH);
    node_prep<<<(nH + 127) / 128, 128, 0, stream>>>(fH, as2, ad2, fAS, fAD, n, H, C);
    edge_logit_max<<<(eH + 255) / 256, 256, 0, stream>>>(ei, E, Et, H, fAS, fAD, fALPH, uMAX);
    edge_exp_sum<<<(eH + 255) / 256, 256, 0, stream>>>(ei, E, Et, H, fALPH, uMAX, fSUM);
    edge_norm<<<(eH + 255) / 256, 256, 0, stream>>>(ei, E, Et, H, fALPH, fSUM);
    edge_scatter<<<scBlocks, 256, 0, stream>>>(ei, E, Et, H, C, fH, fALPH, fACC);
    finalize_residual<<<(n32 + 255) / 256, 256, 0, stream>>>((const float4*)fACC, (const float4*)b2,
                                                             (const float4*)x, (float4*)d_out, n32);
  }
}